// FlashDeformAttnTorch_41601053229312
// MI455X (gfx1250) — compile-verified
//
#include <hip/hip_runtime.h>
#include <hip/hip_bf16.h>
#include <math.h>

// ---------------- problem constants (fixed in the reference module) --------
#define D_MODEL   256
#define N_LEVELS  4
#define N_HEADS   8
#define N_POINTS  4
#define DH        32
#define LEN_IN    11253
#define BATCH     2
#define NQ        LEN_IN
#define MROWS     (BATCH * NQ)          // 22506 rows for every GEMM
#define KMAX      256                   // every GEMM here has K == 256
#define KPSTRIDE  160                   // dwords per k-pair row in LDS (bank-conflict-free)

typedef __attribute__((ext_vector_type(2))) float v2f;
typedef __attribute__((ext_vector_type(8))) float v8f;

// ---------------------------------------------------------------------------
// Generic fp32 WMMA GEMM:  C[M,N] = A[M,K] * W[K,N] + bias[N]   (K == 256)
// Block: 256 threads = 8 waves; block tile 128(M) x 64(N).
//
// W[:, colBase:colBase+64] is staged into LDS once per block with gfx1250
// async-to-LDS (ASYNCcnt), k-pair INTERLEAVED so that each WMMA B fragment
// {W[k'][c], W[k'+1][c]} is one contiguous 8-byte LDS word:
//     lds_dword[(k/2)*KPSTRIDE + 2*c + (k&1)]
// KPSTRIDE=160 dwords => the +half (=+160 dwords = +32 banks) puts lanes
// 16..31 on banks 32..63 while lanes 0..15 use banks 0..31: conflict-free
// ds_load_b64 fragments, landing directly in even-aligned VGPR pairs (WMMA
// requires even-aligned sources - no shuffle movs).
//
// V_WMMA_F32_16X16X4_F32 fragment layouts (ISA 7.12.2):
//   A 16x4 : lane<16 -> (M=lane, K={0,1}) in v[0],v[1]; lane>=16 -> K={2,3}
//   B 4x16 : lane<16 -> (K={0,1}, N=lane); lane>=16 -> K={2,3}
//   C 16x16: vgpr r, lane<16 -> (M=r, N=lane); lane>=16 -> (M=r+8, N=lane-16)
// ---------------------------------------------------------------------------
__global__ __launch_bounds__(256)
void wmma_gemm_bias_f32(const float* __restrict__ A,
                        const float* __restrict__ W,
                        const float* __restrict__ bias,
                        float* __restrict__ C,
                        int M, int N, int K)
{
    __shared__ float wt[(KMAX / 2) * KPSTRIDE];   // 80 KB

    const int lane = threadIdx.x & 31;
    const int wave = threadIdx.x >> 5;      // 0..7
    const int half = lane >> 4;             // 0 or 1
    const int l16  = lane & 15;

    const int rowBase = blockIdx.x * 128 + wave * 16;
    const int colBase = blockIdx.y * 64;

    // ---- async stage W[:, colBase:colBase+64] into interleaved LDS -------
    // One async_b32 instruction moves 32 dwords (one per lane); global side
    // is 128B-contiguous per wave, LDS side performs the k-pair interleave.
    // LDS byte offset = low 32 bits of the flat shared-aperture address
    {
        const unsigned lbase = (unsigned)(uintptr_t)(void*)wt;
        for (int i = 0; i < K / 4; ++i) {                // K*64/256 iters
            const int gidx = threadIdx.x + i * 256;      // 0 .. K*64-1
            const int kk = gidx >> 6;                    // row in [0,K)
            const int cc = gidx & 63;                    // col in [0,64)
            const unsigned long long gaddr =
                (unsigned long long)(uintptr_t)(W + (size_t)kk * N + colBase + cc);
            const unsigned laddr =
                lbase + (unsigned)((kk >> 1) * KPSTRIDE + cc * 2 + (kk & 1)) * 4u;
            asm volatile("global_load_async_to_lds_b32 %0, %1, off"
                         :: "v"(laddr), "v"(gaddr) : "memory");
        }
        asm volatile("s_wait_asynccnt 0x0" ::: "memory");
    }
    __syncthreads();

    // clamp row for loads (WMMA needs EXEC all-1s -> no lane predication);
    // out-of-range rows are masked at the store.
    int aRow = rowBase + l16;
    if (aRow > M - 1) aRow = M - 1;
    const float* Ap = A + (size_t)aRow * K + 2 * half;   // K=2*half,2*half+1

    v8f acc0 = {}, acc1 = {}, acc2 = {}, acc3 = {};

    for (int k = 0; k < K; k += 4) {
        v2f a = *(const v2f*)(Ap + k);                   // 8B-aligned pair
        // B fragments: one ds_load_b64 each, conflict-free banks
        const float* ws = wt + ((k >> 1) + half) * KPSTRIDE + l16 * 2;
        v2f b0 = *(const v2f*)(ws + 0);    // cols j*16 + l16, j = 0..3
        v2f b1 = *(const v2f*)(ws + 32);
        v2f b2 = *(const v2f*)(ws + 64);
        v2f b3 = *(const v2f*)(ws + 96);
        acc0 = __builtin_amdgcn_wmma_f32_16x16x4_f32(false, a, false, b0, (short)0, acc0, false, false);
        acc1 = __builtin_amdgcn_wmma_f32_16x16x4_f32(false, a, false, b1, (short)0, acc1, false, false);
        acc2 = __builtin_amdgcn_wmma_f32_16x16x4_f32(false, a, false, b2, (short)0, acc2, false, false);
        acc3 = __builtin_amdgcn_wmma_f32_16x16x4_f32(false, a, false, b3, (short)0, acc3, false, false);
    }

    const int rowOut = rowBase + 8 * half;
    #pragma unroll
    for (int j = 0; j < 4; ++j) {
        v8f acc = (j == 0) ? acc0 : (j == 1) ? acc1 : (j == 2) ? acc2 : acc3;
        const int col = colBase + j * 16 + l16;
        const float bv = bias[col];
        #pragma unroll
        for (int r = 0; r < 8; ++r) {
            const int row = rowOut + r;
            if (row < M)
                C[(size_t)row * N + col] = acc[r] + bv;
        }
    }
}

// ---------------------------------------------------------------------------
// Deformable sampling: one wave per (b, q, head); lane = channel d (DH==32).
// Softmax over the 16 (level,point) logits, bilinear-gather from the value
// tensor, accumulate the weighted sum into mid[bq, g*32+d].
// All control flow is wave-uniform (location depends only on b,q,g,l,k).
// ---------------------------------------------------------------------------
__global__ __launch_bounds__(256)
void deform_sample_kernel(const float* __restrict__ value,   // (B*LEN_IN, 256)
                          const float* __restrict__ offs,    // (B*Q, 256)
                          const float* __restrict__ logits,  // (B*Q, 128)
                          const float* __restrict__ refpts,  // (B, Q, 4, 2)
                          float* __restrict__ mid)           // (B*Q, 256)
{
    const int cH[4] = {92, 46, 23, 12};
    const int cW[4] = {92, 46, 23, 12};
    const int cS[4] = {0, 8464, 10580, 11109};

    const int lane = threadIdx.x & 31;     // channel d
    const int g    = threadIdx.x >> 5;     // head (8 waves per block)
    const int bq   = blockIdx.x;           // 0 .. B*Q-1
    const int b    = bq / NQ;

    // softmax over 16 logits of this head (wave-uniform loads)
    const float* lg = logits + (size_t)bq * 128 + g * 16;
    float w[16];
    float mx = lg[0];
    #pragma unroll
    for (int i = 1; i < 16; ++i) mx = fmaxf(mx, lg[i]);
    float s = 0.f;
    #pragma unroll
    for (int i = 0; i < 16; ++i) { w[i] = __expf(lg[i] - mx); s += w[i]; }
    const float inv = 1.0f / s;

    const float* of = offs + (size_t)bq * 256 + g * 32;   // (l*4+k)*2 + c
    const float* rp = refpts + (size_t)bq * 8;            // (l, c)

    float acc = 0.f;
    #pragma unroll
    for (int l = 0; l < 4; ++l) {
        const int H = cH[l], W = cW[l];
        const float fH = (float)H, fW = (float)W;
        const float rx = rp[l * 2 + 0], ry = rp[l * 2 + 1];
        const float* vb = value + ((size_t)b * LEN_IN + cS[l]) * 256 + g * 32 + lane;
        #pragma unroll
        for (int k = 0; k < 4; ++k) {
            const float ox = of[(l * 4 + k) * 2 + 0];
            const float oy = of[(l * 4 + k) * 2 + 1];
            const float h_im = (ry + oy / fH) * fH - 0.5f;
            const float w_im = (rx + ox / fW) * fW - 0.5f;
            const float h0f = floorf(h_im), w0f = floorf(w_im);
            const float lh = h_im - h0f, lw = w_im - w0f;
            const float hh = 1.f - lh,   hw = 1.f - lw;
            const int h0 = (int)h0f, w0 = (int)w0f;
            float smp = 0.f;
            if (h0 >= 0 && h0 < H) {
                if (w0 >= 0     && w0 < W)     smp += hh * hw * vb[(size_t)(h0 * W + w0) * 256];
                if (w0 + 1 >= 0 && w0 + 1 < W) smp += hh * lw * vb[(size_t)(h0 * W + w0 + 1) * 256];
            }
            if (h0 + 1 >= 0 && h0 + 1 < H) {
                if (w0 >= 0     && w0 < W)     smp += lh * hw * vb[(size_t)((h0 + 1) * W + w0) * 256];
                if (w0 + 1 >= 0 && w0 + 1 < W) smp += lh * lw * vb[(size_t)((h0 + 1) * W + w0 + 1) * 256];
            }
            const bool valid = (h_im > -1.f) && (w_im > -1.f) && (h_im < fH) && (w_im < fW);
            acc += (valid ? smp : 0.f) * (w[l * 4 + k] * inv);
        }
    }
    mid[(size_t)bq * 256 + g * 32 + lane] = acc;
}

// ---------------------------------------------------------------------------
// in_sizes order (setup_inputs): 0 query, 1 reference_points, 2 input_flatten,
// 3 spatial_shapes(i64), 4 level_start_index(i64), 5 W_off, 6 b_off,
// 7 W_attn, 8 b_attn, 9 W_val, 10 b_val, 11 W_out, 12 b_out.
// Workspace layout (floats): value[M*256] | offsets[M*256] | logits[M*128] |
// mid[M*256]  -> 22506*896*4 B ~= 80.7 MB.
// ---------------------------------------------------------------------------
extern "C" void kernel_launch(void* const* d_in, const int* in_sizes, int n_in,
                              void* d_out, int out_size, void* d_ws, size_t ws_size,
                              hipStream_t stream)
{
    const float* query  = (const float*)d_in[0];
    const float* refpts = (const float*)d_in[1];
    const float* inflat = (const float*)d_in[2];
    const float* W_off  = (const float*)d_in[5];
    const float* b_off  = (const float*)d_in[6];
    const float* W_attn = (const float*)d_in[7];
    const float* b_attn = (const float*)d_in[8];
    const float* W_val  = (const float*)d_in[9];
    const float* b_val  = (const float*)d_in[10];
    const float* W_out  = (const float*)d_in[11];
    const float* b_out  = (const float*)d_in[12];

    float* val_ws = (float*)d_ws;
    float* off_ws = val_ws + (size_t)MROWS * 256;
    float* log_ws = off_ws + (size_t)MROWS * 256;
    float* mid_ws = log_ws + (size_t)MROWS * 128;

    dim3 blk(256);
    dim3 gridN256((MROWS + 127) / 128, 256 / 64);
    dim3 gridN128((MROWS + 127) / 128, 128 / 64);

    // 1) value = input_flatten @ W_val + b_val
    wmma_gemm_bias_f32<<<gridN256, blk, 0, stream>>>(inflat, W_val, b_val, val_ws, MROWS, 256, 256);
    // 2) offsets = query @ W_off + b_off
    wmma_gemm_bias_f32<<<gridN256, blk, 0, stream>>>(query, W_off, b_off, off_ws, MROWS, 256, 256);
    // 3) logits = query @ W_attn + b_attn
    wmma_gemm_bias_f32<<<gridN128, blk, 0, stream>>>(query, W_attn, b_attn, log_ws, MROWS, 128, 256);
    // 4) softmax + bilinear sampling + head-weighted sum
    deform_sample_kernel<<<MROWS, blk, 0, stream>>>(val_ws, off_ws, log_ws, refpts, mid_ws);
    // 5) out = mid @ W_out + b_out
    wmma_gemm_bias_f32<<<gridN256, blk, 0, stream>>>(mid_ws, W_out, b_out, (float*)d_out, MROWS, 256, 256);
}